// SparseConvTransposeBlock_56392920596827
// MI455X (gfx1250) — compile-verified
//
#include <hip/hip_runtime.h>
#include <hip/hip_bf16.h>

typedef __attribute__((ext_vector_type(2))) float v2f;
typedef __attribute__((ext_vector_type(8))) float v8f;

#define C_DIM 64
#define BN_EPS 1e-5f

// ---------------------------------------------------------------------------
// Pass 1: zero the fp32 accumulator (d_out) and the BN stats scratch.
// ---------------------------------------------------------------------------
__global__ void zero_kernel(float* __restrict__ out, int n, float* __restrict__ stats) {
    int stride = gridDim.x * blockDim.x;
    for (int i = blockIdx.x * blockDim.x + threadIdx.x; i < n; i += stride)
        out[i] = 0.0f;
    if (blockIdx.x == 0 && threadIdx.x < 128)
        stats[threadIdx.x] = 0.0f;
}

// ---------------------------------------------------------------------------
// Pass 2: gather -> per-offset GEMM (WMMA f32 16x16x4) -> atomic scatter-add.
// grid.x = k (kernel offset), grid.y covers p-tiles (8 waves / block).
// weight[k] (64x64 fp32, 16KB) is staged TRANSPOSED into LDS so that B
// fragments (K contiguous per N) are v2f ds_load_b64 reads.
// ---------------------------------------------------------------------------
__global__ __launch_bounds__(256) void sparse_conv_wmma(
    const float* __restrict__ features,   // [N_in, 64]
    const float* __restrict__ weight,     // [K3, 64, 64] (c_in, c_out)
    const int*   __restrict__ pairs_in,   // [K3, P]
    const int*   __restrict__ pairs_out,  // [K3, P]
    float*       __restrict__ out,        // [N_out, 64] accumulator
    int P, int tilesPerK)
{
    __shared__ float wT[C_DIM * C_DIM];   // wT[n*64 + c] = W[c*64 + n]

    const int k = blockIdx.x;
    const float* W = weight + (size_t)k * C_DIM * C_DIM;
    for (int i = threadIdx.x; i < C_DIM * C_DIM; i += blockDim.x) {
        int c = i >> 6, n = i & 63;
        wT[n * C_DIM + c] = W[i];
    }
    __syncthreads();

    const int wave = threadIdx.x >> 5;
    const int lane = threadIdx.x & 31;
    const int half = lane >> 4;     // 0: K%4 in {0,1}, 1: K%4 in {2,3}
    const int ml   = lane & 15;     // A row within tile / B,C column within 16

    const int tile = blockIdx.y * 8 + wave;
    if (tile >= tilesPerK) return;
    const int p0 = tile * 16;

    // --- Gather A fragments (16 K-blocks of 16x4 f32; 2 VGPRs each) ---
    const int prow = p0 + ml;
    const bool rowValid = (prow < P);
    const int fidx = rowValid ? pairs_in[(size_t)k * P + prow] : 0;
    const float* frow = features + (size_t)fidx * C_DIM + half * 2;

    v2f a[16];
#pragma unroll
    for (int kb = 0; kb < 16; ++kb) {
        v2f t = *(const v2f*)(frow + kb * 4);
        if (!rowValid) { t.x = 0.0f; t.y = 0.0f; }
        a[kb] = t;
    }

    // --- Output row indices for the scatter (C layout: VGPR v -> M=v+half*8) ---
    int orow[8];
#pragma unroll
    for (int v = 0; v < 8; ++v) {
        int pp = p0 + v + half * 8;
        orow[v] = (pp < P) ? pairs_out[(size_t)k * P + pp] : -1;
    }

    // --- 4 column tiles of 16; accumulate over K=64 in steps of 4 ---
#pragma unroll
    for (int nb = 0; nb < 4; ++nb) {
        v8f acc = {};
        const float* bt = &wT[(nb * 16 + ml) * C_DIM + half * 2];
#pragma unroll
        for (int kb = 0; kb < 16; ++kb) {
            v2f b = *(const v2f*)(bt + kb * 4);
            acc = __builtin_amdgcn_wmma_f32_16x16x4_f32(
                false, a[kb], false, b, (short)0, acc, false, false);
        }
        const int col = nb * 16 + ml;
#pragma unroll
        for (int v = 0; v < 8; ++v) {
            if (orow[v] >= 0)
                unsafeAtomicAdd(&out[(size_t)orow[v] * C_DIM + col], acc[v]);
        }
    }
}

// ---------------------------------------------------------------------------
// Pass 3: per-channel sum and sum-of-squares partials -> fp32 global atomics.
// ---------------------------------------------------------------------------
__global__ __launch_bounds__(256) void bn_stats(
    const float* __restrict__ out, int nrows, float* __restrict__ stats)
{
    const int c  = threadIdx.x & 63;
    const int rg = threadIdx.x >> 6;   // 0..3 row groups per block
    float s = 0.0f, s2 = 0.0f;
    for (int r = blockIdx.x * 4 + rg; r < nrows; r += gridDim.x * 4) {
        float x = out[(size_t)r * C_DIM + c];
        s  += x;
        s2 += x * x;
    }
    __shared__ float sh[2][4][C_DIM];
    sh[0][rg][c] = s;
    sh[1][rg][c] = s2;
    __syncthreads();
    if (rg == 0) {
        float ts  = sh[0][0][c] + sh[0][1][c] + sh[0][2][c] + sh[0][3][c];
        float ts2 = sh[1][0][c] + sh[1][1][c] + sh[1][2][c] + sh[1][3][c];
        unsafeAtomicAdd(&stats[c], ts);
        unsafeAtomicAdd(&stats[C_DIM + c], ts2);
    }
}

// ---------------------------------------------------------------------------
// Pass 4: fold stats into per-channel scale/shift.
// (Conv bias cancels exactly inside BN: mean shifts by the same constant.)
// ---------------------------------------------------------------------------
__global__ void bn_params(const float* __restrict__ stats,
                          const float* __restrict__ gamma,
                          const float* __restrict__ beta,
                          float* __restrict__ params, int nrows)
{
    int c = threadIdx.x;
    if (c >= C_DIM) return;
    float inv_n = 1.0f / (float)nrows;
    float mean  = stats[c] * inv_n;
    float var   = stats[C_DIM + c] * inv_n - mean * mean;
    float scale = gamma[c] * rsqrtf(var + BN_EPS);
    params[c]         = scale;
    params[C_DIM + c] = beta[c] - mean * scale;
}

// ---------------------------------------------------------------------------
// Pass 5: in-place normalize + ReLU.
// ---------------------------------------------------------------------------
__global__ void bn_relu(float* __restrict__ out, int n,
                        const float* __restrict__ params)
{
    int stride = gridDim.x * blockDim.x;
    for (int i = blockIdx.x * blockDim.x + threadIdx.x; i < n; i += stride) {
        int c = i & 63;
        float y = fmaf(out[i], params[c], params[C_DIM + c]);
        out[i] = fmaxf(y, 0.0f);
    }
}

// ---------------------------------------------------------------------------
extern "C" void kernel_launch(void* const* d_in, const int* in_sizes, int n_in,
                              void* d_out, int out_size, void* d_ws, size_t ws_size,
                              hipStream_t stream) {
    const float* features  = (const float*)d_in[0];
    const float* weight    = (const float*)d_in[1];
    // d_in[2] = bias (cancels inside BatchNorm -> exact to omit)
    const float* gamma     = (const float*)d_in[3];
    const float* beta      = (const float*)d_in[4];
    const int*   pairs_in  = (const int*)d_in[5];
    const int*   pairs_out = (const int*)d_in[6];
    float* out = (float*)d_out;

    const int K3    = in_sizes[1] / (C_DIM * C_DIM);   // 27
    const int P     = in_sizes[5] / K3;                // 80000
    const int n_out = out_size / C_DIM;                // 200000

    float* stats  = (float*)d_ws;       // [128]: sum, sumsq
    float* params = stats + 128;        // [128]: scale, shift

    zero_kernel<<<1024, 256, 0, stream>>>(out, out_size, stats);

    const int tilesPerK = (P + 15) / 16;
    dim3 grid(K3, (tilesPerK + 7) / 8);
    sparse_conv_wmma<<<grid, 256, 0, stream>>>(features, weight, pairs_in,
                                               pairs_out, out, P, tilesPerK);

    bn_stats<<<512, 256, 0, stream>>>(out, n_out, stats);
    bn_params<<<1, 64, 0, stream>>>(stats, gamma, beta, params, n_out);
    bn_relu<<<1024, 256, 0, stream>>>(out, out_size, params);
}